// Relation_51015621542235
// MI455X (gfx1250) — compile-verified
//
#include <hip/hip_runtime.h>
#include <cstdint>

#define HDIM 48
#define WDIM 48
#define PIX  (HDIM * WDIM)   // 2304
#define CIN  512
#define NB   4
#define NHEADS 8
#define HD   64
#define KK   25

typedef float v2f __attribute__((ext_vector_type(2)));
typedef float v8f __attribute__((ext_vector_type(8)));

// ---------------------------------------------------------------------------
// GEMM: Y[n][m][hw] = sum_c Wm[m][c] * X[n][c][hw]   (fp32 via WMMA 16x16x4)
// Block = 256 threads (8 waves). Each block owns 32 pixels (two 16-pixel
// tiles) and 128 output rows (16 per wave). Each wave shares one A fragment
// across two B fragments -> two independent WMMA accumulation chains.
// grid.x = (NB*PIX)/32 pixel tiles, grid.y = Mtot/128
// ---------------------------------------------------------------------------
__global__ __launch_bounds__(256)
void gemm512_wmma(const float* __restrict__ Wm,
                  const float* __restrict__ X,
                  float* __restrict__ Y,
                  int Mtot)
{
    // Two separate [c][16] tiles keep the bank-conflict-free access pattern.
    __shared__ float xs[2][CIN][16];   // 64 KB

    const int gp  = blockIdx.x * 32;     // global pixel base (contiguous in hw)
    const int n   = gp / PIX;
    const int hw0 = gp % PIX;

    // Stage 32-pixel x 512-channel input tile into LDS with b128 loads.
    {
        const float* xb = X + ((size_t)n * CIN) * PIX + hw0;
        const int p4 = (threadIdx.x & 7) * 4;   // pixel quad within 32
        const int c0 = threadIdx.x >> 3;        // 0..31
        const int j  = p4 >> 4;                 // which 16-pixel tile
        const int pl = p4 & 15;                 // pixel within tile
        #pragma unroll
        for (int i = 0; i < CIN / 32; ++i) {
            const int c = c0 + i * 32;
            *(float4*)&xs[j][c][pl] = *(const float4*)(xb + (size_t)c * PIX + p4);
        }
    }
    __syncthreads();

    const int lane = threadIdx.x & 31;
    const int wv   = threadIdx.x >> 5;
    const int lm   = lane & 15;
    const int hi   = lane >> 4;      // half-wave select
    const int koff = hi << 1;        // K offset 0 or 2 within the 16x4 A tile

    const int dBase = (blockIdx.y * 8 + wv) * 16;
    const float* wrow = Wm + (size_t)(dBase + lm) * CIN + koff;

    v8f acc0 = {0.f, 0.f, 0.f, 0.f, 0.f, 0.f, 0.f, 0.f};
    v8f acc1 = {0.f, 0.f, 0.f, 0.f, 0.f, 0.f, 0.f, 0.f};

    #pragma unroll 8
    for (int k = 0; k < CIN / 4; ++k) {
        // A fragment: lane L holds W[dBase+L%16][4k+koff .. +1]  (8B aligned)
        v2f a = *(const v2f*)(wrow + 4 * k);
        // B fragments: lane L holds X[4k+koff .. +1][L%16] for each tile
        v2f b0, b1;
        b0.x = xs[0][4 * k + koff][lm];
        b0.y = xs[0][4 * k + koff + 1][lm];
        b1.x = xs[1][4 * k + koff][lm];
        b1.y = xs[1][4 * k + koff + 1][lm];
        acc0 = __builtin_amdgcn_wmma_f32_16x16x4_f32(
            false, a, false, b0, (short)0, acc0, false, false);
        acc1 = __builtin_amdgcn_wmma_f32_16x16x4_f32(
            false, a, false, b1, (short)0, acc1, false, false);
    }

    // D layout: VGPR r -> row (r + 8*hi), col lm
    float* yp = Y + ((size_t)n * Mtot + dBase + hi * 8) * PIX + hw0 + lm;
    #pragma unroll
    for (int r = 0; r < 8; ++r)
        yp[(size_t)r * PIX] = acc0[r];
    #pragma unroll
    for (int r = 0; r < 8; ++r)
        yp[(size_t)r * PIX + 16] = acc1[r];
}

// ---------------------------------------------------------------------------
// Neighborhood attention: one thread per (n, head, pixel).
// qkv layout: [n][d][hw], d: Q=[0,512) K=[512,1024) V=[1024,1536)
// ---------------------------------------------------------------------------
__global__ __launch_bounds__(256)
void natt_kernel(const float* __restrict__ qkv, float* __restrict__ attn)
{
    const int t  = blockIdx.x * 256 + threadIdx.x;   // < 4*8*2304 = 73728
    const int hw = t % PIX;
    const int g  = (t / PIX) & (NHEADS - 1);
    const int n  = t / (PIX * NHEADS);
    const int h  = hw / WDIM;
    const int w  = hw % WDIM;

    const size_t nbase = (size_t)n * (3 * CIN) * PIX;
    const float* Qp = qkv + nbase + (size_t)(g * HD) * PIX + hw;
    const float* Kp = Qp + (size_t)CIN * PIX;
    const float* Vp = Qp + (size_t)(2 * CIN) * PIX;

    // Zero-padded patches: OOB neighbors contribute score 0 (still in softmax)
    // and contribute nothing to the V-sum.
    float mask[KK];
    #pragma unroll
    for (int k = 0; k < KK; ++k) {
        const int dy = k / 5 - 2, dx = k % 5 - 2;
        mask[k] = (h + dy >= 0 && h + dy < HDIM &&
                   w + dx >= 0 && w + dx < WDIM) ? 1.f : 0.f;
    }

    float s[KK];
    #pragma unroll
    for (int k = 0; k < KK; ++k) s[k] = 0.f;

    for (int c = 0; c < HD; ++c) {
        const float q = Qp[(size_t)c * PIX];
        const float* kc = Kp + (size_t)c * PIX;
        #pragma unroll
        for (int k = 0; k < KK; ++k) {
            const int off = (k / 5 - 2) * WDIM + (k % 5 - 2);
            s[k] = fmaf(mask[k] * kc[off], q, s[k]);
        }
    }

    // softmax over all 25 (OOB entries are exactly 0, matching the reference)
    float m = s[0];
    #pragma unroll
    for (int k = 1; k < KK; ++k) m = fmaxf(m, s[k]);
    float sum = 0.f;
    #pragma unroll
    for (int k = 0; k < KK; ++k) { s[k] = __expf(s[k] - m); sum += s[k]; }
    const float inv = 1.f / sum;
    #pragma unroll
    for (int k = 0; k < KK; ++k) s[k] *= inv * mask[k];  // mask V contributions

    float* Op = attn + ((size_t)n * CIN + g * HD) * PIX + hw;
    for (int c = 0; c < HD; ++c) {
        const float* vc = Vp + (size_t)c * PIX;
        float acc = 0.f;
        #pragma unroll
        for (int k = 0; k < KK; ++k) {
            const int off = (k / 5 - 2) * WDIM + (k % 5 - 2);
            acc = fmaf(s[k], vc[off], acc);
        }
        Op[(size_t)c * PIX] = acc;
    }
}

// ---------------------------------------------------------------------------
extern "C" void kernel_launch(void* const* d_in, const int* in_sizes, int n_in,
                              void* d_out, int out_size, void* d_ws, size_t ws_size,
                              hipStream_t stream)
{
    const float* x     = (const float*)d_in[0];   // (4, 512, 48, 48)
    const float* w_qkv = (const float*)d_in[1];   // (1536, 512)
    const float* w_t   = (const float*)d_in[2];   // (512, 512)
    float*       out   = (float*)d_out;           // (4, 512, 48, 48)

    float* qkvbuf  = (float*)d_ws;                          // 4*1536*2304 floats
    float* attnbuf = qkvbuf + (size_t)NB * (3 * CIN) * PIX; // 4*512*2304 floats

    // 1) QKV projection: (1536x512) x (512 x 9216)
    gemm512_wmma<<<dim3((NB * PIX) / 32, (3 * CIN) / 128), 256, 0, stream>>>(
        w_qkv, x, qkvbuf, 3 * CIN);

    // 2) Neighborhood attention
    natt_kernel<<<(NB * NHEADS * PIX) / 256, 256, 0, stream>>>(qkvbuf, attnbuf);

    // 3) Output projection: (512x512) x (512 x 9216)
    gemm512_wmma<<<dim3((NB * PIX) / 32, CIN / 128), 256, 0, stream>>>(
        w_t, attnbuf, out, CIN);
}